// CensusLoss_28260884807724
// MI455X (gfx1250) — compile-verified
//
#include <hip/hip_runtime.h>
#include <hip/hip_bf16.h>
#include <stdint.h>

#define IMG_H 512
#define IMG_W 512
#define BATCH 8
#define PATCH 7
#define PADR  3
#define TILE  32
#define LDSH  (TILE + 2*PADR)   /* 38 */
#define LDSW  40                /* padded row stride: conflict-free */
#define EPSC  1e-6f
#define NBLOCKS 2048            /* (512/32)*(512/32)*8 */

typedef __attribute__((ext_vector_type(2))) float v2f;
typedef __attribute__((ext_vector_type(8))) float v8f;

#if __has_builtin(__builtin_amdgcn_wmma_f32_16x16x4_f32)
#define HAVE_WMMA_F32X4 1
#endif

#if __has_builtin(__builtin_amdgcn_global_load_async_to_lds_b32)
#define HAVE_ASYNC_LDS 1
typedef __attribute__((address_space(1))) int gas_int;   // global AS pointee
typedef __attribute__((address_space(3))) int las_int;   // LDS AS pointee
#endif

// Sum of one float per lane across the full wave32.
// WMMA path: A[m,0]=v_m (lanes 0-15), A[m,2]=v_{m+16} (lanes 16-31), rest 0.
// B = ones => D[m,n] = v_m + v_{m+16}. Lanes 0-15 hold D rows 0..7 in 8 VGPRs,
// lanes 16-31 hold rows 8..15; sum the 8 regs + shfl_xor(16) => total.
__device__ __forceinline__ float wave_sum32(float v) {
#ifdef HAVE_WMMA_F32X4
  v2f a; a.x = v;    a.y = 0.0f;
  v2f b; b.x = 1.0f; b.y = 1.0f;
  v8f c = {};
  v8f d = __builtin_amdgcn_wmma_f32_16x16x4_f32(false, a, false, b,
                                                (short)0, c, false, false);
  float s = ((d[0]+d[1])+(d[2]+d[3])) + ((d[4]+d[5])+(d[6]+d[7]));
  s += __shfl_xor(s, 16, 32);
  return s;
#else
  #pragma unroll
  for (int off = 16; off; off >>= 1) v += __shfl_xor(v, off, 32);
  return v;
#endif
}

// Pass 1: diff = gray(pred) - gray(gt), vectorized float4. 524288 threads.
__global__ __launch_bounds__(256) void census_diff_kernel(
    const float* __restrict__ pred, const float* __restrict__ gt,
    float* __restrict__ diff) {
  const int plane4 = (IMG_H * IMG_W) / 4;          // 65536
  int i = blockIdx.x * 256 + threadIdx.x;          // float4 index
  int b = i / plane4;
  int p = i - b * plane4;
  const float4* pr = (const float4*)pred;
  const float4* gr = (const float4*)gt;
  long base = (long)b * 3 * plane4 + p;
  float4 p0 = pr[base], p1 = pr[base + plane4], p2 = pr[base + 2*plane4];
  float4 g0 = gr[base], g1 = gr[base + plane4], g2 = gr[base + 2*plane4];
  float4 o;
  o.x = 0.299f*(p0.x-g0.x) + 0.587f*(p1.x-g1.x) + 0.114f*(p2.x-g2.x);
  o.y = 0.299f*(p0.y-g0.y) + 0.587f*(p1.y-g1.y) + 0.114f*(p2.y-g2.y);
  o.z = 0.299f*(p0.z-g0.z) + 0.587f*(p1.z-g1.z) + 0.114f*(p2.z-g2.z);
  o.w = 0.299f*(p0.w-g0.w) + 0.587f*(p1.w-g1.w) + 0.114f*(p2.w-g2.w);
  ((float4*)diff)[i] = o;
}

// Pass 2: 32x32 tile per block (+3 halo each side, zero-padded), 7x7 census
// charbonnier sum per pixel, block partial to ws. Tile staged via async
// global->LDS (ASYNCcnt path) when available.
__global__ __launch_bounds__(256) void census_tile_kernel(
    const float* __restrict__ diff, float* __restrict__ partials) {
  __shared__ float tile[LDSH * LDSW];
  __shared__ float wsums[8];
  const int tid = threadIdx.x;
  const int bx = blockIdx.x, by = blockIdx.y, bz = blockIdx.z;
  const int gx0 = bx * TILE - PADR;
  const int gy0 = by * TILE - PADR;
  const float* dplane = diff + (size_t)bz * (IMG_H * IMG_W);

  // zero-fill covers out-of-image halo (zero padding semantics)
  for (int e = tid; e < LDSH * LDSW; e += 256) tile[e] = 0.0f;
  __syncthreads();

  for (int e = tid; e < LDSH * LDSH; e += 256) {
    int ly = e / LDSH;
    int lx = e - ly * LDSH;
    int gy = gy0 + ly, gx = gx0 + lx;
    if ((unsigned)gy < IMG_H && (unsigned)gx < IMG_W) {
#ifdef HAVE_ASYNC_LDS
      __builtin_amdgcn_global_load_async_to_lds_b32(
          (gas_int*)(dplane + gy * IMG_W + gx),
          (las_int*)&tile[ly * LDSW + lx], 0, 0);
#else
      tile[ly * LDSW + lx] = dplane[gy * IMG_W + gx];
#endif
    }
  }
#ifdef HAVE_ASYNC_LDS
#if __has_builtin(__builtin_amdgcn_s_wait_asynccnt)
  __builtin_amdgcn_s_wait_asynccnt(0);
#else
  asm volatile("s_wait_asynccnt 0" ::: "memory");
#endif
#endif
  __syncthreads();

  const int tx  = tid & 31;   // lanes map to consecutive x -> conflict-free
  const int ty0 = tid >> 5;   // 4 rows per thread, stride 8
  float acc = 0.0f;
  #pragma unroll
  for (int i = 0; i < 4; ++i) {
    int y = ty0 + (i << 3);
    float cv = tile[(y + PADR) * LDSW + (tx + PADR)];
    #pragma unroll
    for (int dy = 0; dy < PATCH; ++dy) {
      int rb = (y + dy) * LDSW + tx;
      #pragma unroll
      for (int dx = 0; dx < PATCH; ++dx) {
        float d = tile[rb + dx] - cv;
        acc += __builtin_amdgcn_sqrtf(__builtin_fmaf(d, d, EPSC));
      }
    }
  }

  // wave reduce then cross-wave via WMMA ones-trick
  float a = acc;
  #pragma unroll
  for (int off = 16; off; off >>= 1) a += __shfl_down(a, off, 32);
  if ((tid & 31) == 0) wsums[tid >> 5] = a;
  __syncthreads();
  if (tid < 32) {                       // whole wave 0: EXEC all ones
    float v = (tid < 8) ? wsums[tid] : 0.0f;
    float s = wave_sum32(v);
    if (tid == 0) {
      int bid = (bz * gridDim.y + by) * gridDim.x + bx;
      partials[bid] = s;
    }
  }
}

// Pass 3: reduce 2048 block partials, scale, write scalar.
__global__ __launch_bounds__(256) void census_final_kernel(
    const float* __restrict__ partials, float* __restrict__ out) {
  __shared__ float ls[256];
  const int tid = threadIdx.x;
  float p = 0.0f;
  for (int j = tid; j < NBLOCKS; j += 256) p += partials[j];
  ls[tid] = p;
  __syncthreads();
  if (tid < 32) {
    float v = 0.0f;
    #pragma unroll
    for (int k = 0; k < 8; ++k) v += ls[tid + (k << 5)];
    float s = wave_sum32(v);
    if (tid == 0)
      out[0] = s * (float)(1.0 / ((double)BATCH * PATCH * PATCH * IMG_H * IMG_W));
  }
}

extern "C" void kernel_launch(void* const* d_in, const int* in_sizes, int n_in,
                              void* d_out, int out_size, void* d_ws, size_t ws_size,
                              hipStream_t stream) {
  const float* pred = (const float*)d_in[0];
  const float* gt   = (const float*)d_in[1];
  float* out  = (float*)d_out;
  float* diff = (float*)d_ws;                       // 2,097,152 floats (8 MB)
  float* partials = diff + BATCH * IMG_H * IMG_W;   // 2048 floats

  census_diff_kernel<<<2048, 256, 0, stream>>>(pred, gt, diff);
  census_tile_kernel<<<dim3(IMG_W / TILE, IMG_H / TILE, BATCH), 256, 0, stream>>>(diff, partials);
  census_final_kernel<<<1, 256, 0, stream>>>(partials, out);
}